// Atomistic_27839978013279
// MI455X (gfx1250) — compile-verified
//
#include <hip/hip_runtime.h>

// ---------------------------------------------------------------------------
// MoE per-species linear + segment-sum for MI455X (gfx1250, wave32, WMMA).
//   out[seg] += x[atom] @ W[species[atom]] + b[species[atom]]
// Pipeline: stable counting-sort by species -> 32-row gathered-tile WMMA
// f32 16x16x4 GEMM (B pre-packed to lane layout: 1 global_load_b64 per B
// operand, reused by 2 WMMAs) -> run-length-merged float atomics.
// ---------------------------------------------------------------------------

typedef __attribute__((ext_vector_type(2))) float v2f;
typedef __attribute__((ext_vector_type(8))) float v8f;

#define DIM 256          // D_IN == D_OUT == 256
#define NSPEC 4
#define CHUNK 2048
#define MROWS 32         // atom rows per block (two 16-row WMMA tiles)

// ---------------- zero output ----------------
__global__ void k_zero(float* __restrict__ out, int n) {
    int i = blockIdx.x * blockDim.x + threadIdx.x;
    if (i < n) out[i] = 0.0f;
}

// ---------------- per-chunk species histogram ----------------
__global__ void k_chunk_hist(const int* __restrict__ sp, int n,
                             int* __restrict__ chunkCnt) {
    __shared__ int cnt[NSPEC];
    int tid = threadIdx.x;
    if (tid < NSPEC) cnt[tid] = 0;
    __syncthreads();
    int base = blockIdx.x * CHUNK;
    for (int o = tid; o < CHUNK; o += blockDim.x) {
        int i = base + o;
        if (i < n) atomicAdd(&cnt[sp[i]], 1);
    }
    __syncthreads();
    if (tid < NSPEC) chunkCnt[blockIdx.x * NSPEC + tid] = cnt[tid];
}

// ---------------- serial scan (tiny: nchunk*4 entries) ----------------
__global__ void k_scan(int* __restrict__ counts, int* __restrict__ offsets,
                       int* __restrict__ chunkCnt, int nchunk) {
    if (threadIdx.x != 0 || blockIdx.x != 0) return;
    int tot[NSPEC];
    for (int s = 0; s < NSPEC; ++s) tot[s] = 0;
    for (int c = 0; c < nchunk; ++c)
        for (int s = 0; s < NSPEC; ++s) tot[s] += chunkCnt[c * NSPEC + s];
    int off = 0;
    for (int s = 0; s < NSPEC; ++s) {
        counts[s]  = tot[s];
        offsets[s] = off;
        off += tot[s];
    }
    for (int s = 0; s < NSPEC; ++s) {
        int run = offsets[s];
        for (int c = 0; c < nchunk; ++c) {
            int t = chunkCnt[c * NSPEC + s];
            chunkCnt[c * NSPEC + s] = run;
            run += t;
        }
    }
}

// ---------------- stable scatter: perm sorted by (species, atom id) --------
__global__ void k_scatter(const int* __restrict__ sp, int n,
                          const int* __restrict__ chunkBase,
                          int* __restrict__ perm) {
    __shared__ int runPos[NSPEC];
    __shared__ int waveTot[8][NSPEC];
    __shared__ int waveBase[8][NSPEC];
    int tid  = threadIdx.x;            // 256 threads = 8 waves
    int wid  = tid >> 5;
    int lane = tid & 31;
    int base = blockIdx.x * CHUNK;
    if (tid < NSPEC) runPos[tid] = chunkBase[blockIdx.x * NSPEC + tid];
    __syncthreads();

    for (int step = 0; step < CHUNK; step += 256) {
        int i = base + step + tid;
        int s = (i < n) ? sp[i] : -1;
        unsigned long long lt = (1ull << lane) - 1ull;
        int myRank = 0;
        int tot[NSPEC];
        #pragma unroll
        for (int q = 0; q < NSPEC; ++q) {
            unsigned long long bal = __ballot(s == q);
            tot[q] = __popcll(bal);
            if (s == q) myRank = __popcll(bal & lt);
        }
        if (lane < NSPEC) waveTot[wid][lane] = tot[lane];
        __syncthreads();
        if (tid < NSPEC) {
            int run = 0;
            for (int w = 0; w < 8; ++w) {
                waveBase[w][tid] = run;
                run += waveTot[w][tid];
            }
            waveTot[0][tid] = run;  // step total
        }
        __syncthreads();
        if (s >= 0) perm[runPos[s] + waveBase[wid][s] + myRank] = i;
        __syncthreads();
        if (tid < NSPEC) runPos[tid] += waveTot[0][tid];
        __syncthreads();
    }
}

// ---------------- pack W into WMMA B lane layout -------------------------
// Wp[s][p][c] = float2{ W[s][2p][c], W[s][2p+1][c] }, p in [0,128), c in [0,256)
__global__ void k_repack(const float* __restrict__ W, float* __restrict__ Wp,
                         int total /* nspec*128*256 */) {
    int i = blockIdx.x * blockDim.x + threadIdx.x;
    if (i >= total) return;
    int c  = i & (DIM - 1);
    int p  = (i >> 8) & 127;
    int s  = i >> 15;
    const float* ws = W + (size_t)s * DIM * DIM;
    Wp[(size_t)i * 2 + 0] = ws[(2 * p + 0) * DIM + c];
    Wp[(size_t)i * 2 + 1] = ws[(2 * p + 1) * DIM + c];
}

// ---------------- main: 32-row gathered-tile WMMA GEMM -------------------
// grid = (ceil(n/32), NSPEC), block = 128 (4 waves). Each wave owns a
// 32x64 output slab: 8 v8f accumulators; each B operand feeds 2 WMMAs.
template <bool PACKED>
__global__ void __launch_bounds__(128)
k_moe(const float* __restrict__ x, const float* __restrict__ Wsrc,
      const float* __restrict__ bias, const int* __restrict__ segs,
      const int* __restrict__ perm, const int* __restrict__ offsets,
      const int* __restrict__ counts, float* __restrict__ out) {
    const int s = blockIdx.y;
    const int cnt = counts[s];
    const int tileBase = blockIdx.x * MROWS;
    if (tileBase >= cnt) return;                 // block-uniform exit
    int valid = cnt - tileBase;
    if (valid > MROWS) valid = MROWS;
    const int base = offsets[s] + tileBase;

    __shared__ float At[MROWS][DIM + 4];         // stride 260: conflict free
    __shared__ int   rowSeg[MROWS];

    const int tid = threadIdx.x;
    if (tid < MROWS) {
        int a = (tid < valid) ? perm[base + tid] : -1;
        rowSeg[tid] = (a >= 0) ? segs[a] : -1;
    }
    // cooperative gather of the 32x256 fp32 A-tile
    for (int i = tid; i < MROWS * (DIM / 4); i += 128) {
        int row = i >> 6;
        int q   = i & 63;
        float4 v = make_float4(0.f, 0.f, 0.f, 0.f);
        if (row < valid) {
            int a = perm[base + row];
            v = *((const float4*)(x + (size_t)a * DIM) + q);
        }
        At[row][q * 4 + 0] = v.x;
        At[row][q * 4 + 1] = v.y;
        At[row][q * 4 + 2] = v.z;
        At[row][q * 4 + 3] = v.w;
    }
    __syncthreads();

    const int wid  = tid >> 5;
    const int lane = tid & 31;
    const int half = lane >> 4;
    const int nn   = lane & 15;
    const int colbase = wid * 64;

    v8f acc[2][4] = {};
    #pragma unroll 2
    for (int k0 = 0; k0 < DIM; k0 += 4) {
        const int kk = k0 + half * 2;            // A/B VGPR0=(K0,K2), VGPR1=(K1,K3)
        v2f a0, a1;
        a0.x = At[nn][kk];       a0.y = At[nn][kk + 1];
        a1.x = At[16 + nn][kk];  a1.y = At[16 + nn][kk + 1];
        #pragma unroll
        for (int t = 0; t < 4; ++t) {
            const int col = colbase + t * 16 + nn;
            v2f b;
            if (PACKED) {
                // packed: one b64 load per B operand
                const float* wp = Wsrc +
                    (((size_t)s * 128 + (size_t)(kk >> 1)) * DIM + col) * 2;
                b.x = wp[0];
                b.y = wp[1];
            } else {
                const float* ws = Wsrc + (size_t)s * DIM * DIM;
                b.x = ws[(size_t)kk * DIM + col];
                b.y = ws[(size_t)(kk + 1) * DIM + col];
            }
            acc[0][t] = __builtin_amdgcn_wmma_f32_16x16x4_f32(
                false, a0, false, b, (short)0, acc[0][t], false, false);
            acc[1][t] = __builtin_amdgcn_wmma_f32_16x16x4_f32(
                false, a1, false, b, (short)0, acc[1][t], false, false);
        }
    }

    // epilogue: bias + run-length merge of consecutive equal segments
    #pragma unroll
    for (int mt = 0; mt < 2; ++mt) {
        #pragma unroll
        for (int t = 0; t < 4; ++t) {
            const int col = colbase + t * 16 + nn;
            const float bv = bias[s * DIM + col];
            int curSeg = -1;
            float sum = 0.0f;
            #pragma unroll
            for (int r = 0; r < 8; ++r) {
                const int M  = mt * 16 + r + 8 * half;
                const int sg = rowSeg[M];
                if (sg >= 0) {
                    const float v = acc[mt][t][r] + bv;
                    if (sg != curSeg) {
                        if (curSeg >= 0)
                            atomicAdd(&out[(size_t)curSeg * DIM + col], sum);
                        curSeg = sg;
                        sum = v;
                    } else {
                        sum += v;
                    }
                }
            }
            if (curSeg >= 0) atomicAdd(&out[(size_t)curSeg * DIM + col], sum);
        }
    }
}

// ---------------------------------------------------------------------------
extern "C" void kernel_launch(void* const* d_in, const int* in_sizes, int n_in,
                              void* d_out, int out_size, void* d_ws, size_t ws_size,
                              hipStream_t stream) {
    const float* x    = (const float*)d_in[0];
    const float* W    = (const float*)d_in[1];
    const float* b    = (const float*)d_in[2];
    const int*   sp   = (const int*)d_in[3];
    const int*   segs = (const int*)d_in[4];
    float* out = (float*)d_out;

    const int n      = in_sizes[3];                 // N_ATOMS
    const int nspec  = in_sizes[1] / (DIM * DIM);   // 4
    const int nchunk = (n + CHUNK - 1) / CHUNK;

    int* wsI      = (int*)d_ws;
    int* counts   = wsI;                         // [4]
    int* offsets  = wsI + 4;                     // [4]
    int* chunkCnt = wsI + 8;                     // [nchunk*4] -> chunk bases
    int* perm     = wsI + 8 + nchunk * NSPEC;    // [n]
    const size_t permEndInts = (size_t)(8 + nchunk * NSPEC + ((n + 1) & ~1));
    float* Wp = (float*)((char*)d_ws + permEndInts * sizeof(int)); // 8B aligned
    const size_t needPacked =
        permEndInts * sizeof(int) + (size_t)nspec * DIM * DIM * sizeof(float);
    (void)n_in;

    k_zero<<<(out_size + 255) / 256, 256, 0, stream>>>(out, out_size);
    k_chunk_hist<<<nchunk, 256, 0, stream>>>(sp, n, chunkCnt);
    k_scan<<<1, 32, 0, stream>>>(counts, offsets, chunkCnt, nchunk);
    k_scatter<<<nchunk, 256, 0, stream>>>(sp, n, chunkCnt, perm);

    dim3 grid((n + MROWS - 1) / MROWS, nspec);
    if (ws_size >= needPacked) {
        const int total = nspec * 128 * DIM;
        k_repack<<<(total + 255) / 256, 256, 0, stream>>>(W, Wp, total);
        k_moe<true><<<grid, 128, 0, stream>>>(x, Wp, b, segs, perm, offsets,
                                              counts, out);
    } else {
        k_moe<false><<<grid, 128, 0, stream>>>(x, W, b, segs, perm, offsets,
                                               counts, out);
    }
}